// BaseAstroGNN_51247549776507
// MI455X (gfx1250) — compile-verified
//
#include <hip/hip_runtime.h>
#include <stdint.h>

#define N_NODES 100000
#define N_EDGES 1600000
#define HID     128
#define IN_DIM  16
#define LN_EPS  1e-5f

typedef __attribute__((ext_vector_type(2))) float v2f;
typedef __attribute__((ext_vector_type(8))) float v8f;

__device__ __forceinline__ v8f wmma_f32_k4(v2f a, v2f b, v8f c) {
  // D = A(16x4,f32) * B(4x16,f32) + C(16x16,f32)
  return __builtin_amdgcn_wmma_f32_16x16x4_f32(
      /*neg_a=*/false, a, /*neg_b=*/false, b,
      /*c_mod=*/(short)0, c, /*reuse_a=*/false, /*reuse_b=*/false);
}

// ---------------- degree / normalization ----------------

__global__ void deg_init_kernel(float* deg, int n) {
  int i = blockIdx.x * blockDim.x + threadIdx.x;
  if (i < n) deg[i] = 1.0f;  // self-loop
}

__global__ void deg_edges_kernel(const long long* __restrict__ dst, float* deg, int e) {
  int i = blockIdx.x * blockDim.x + threadIdx.x;
  if (i < e) atomicAdd(&deg[(int)dst[i]], 1.0f);
}

__global__ void deg_to_dis_kernel(float* deg, int n) {
  int i = blockIdx.x * blockDim.x + threadIdx.x;
  if (i < n) deg[i] = rsqrtf(fmaxf(deg[i], 1.0f));
}

// ---------------- WMMA GEMM: out[nrows,128] = A[nrows,KD] @ W[KD,128] (+bias) ----------------

template <int KD>
__global__ void __launch_bounds__(256) gemm_rowslab_kernel(
    const float* __restrict__ A, const float* __restrict__ W,
    const float* __restrict__ bias, float* __restrict__ out, int nrows) {
  constexpr int AS = KD + 4;           // padded LDS row stride (floats)
  __shared__ float hs[64 * AS];
  const int tid = threadIdx.x;
  const int rbase = blockIdx.x * 64;

  // stage 64 x KD row slab of A into LDS (zero-pad past nrows)
  for (int idx = tid; idx < 64 * KD; idx += 256) {
    int r = idx / KD, c = idx - r * KD;
    int gr = rbase + r;
    hs[r * AS + c] = (gr < nrows) ? A[(size_t)gr * KD + c] : 0.0f;
  }
  __syncthreads();

  const int wave = tid >> 5;           // 8 waves -> 8 column strips of 16
  const int lane = tid & 31;
  const int n  = lane & 15;            // col within strip / A row within tile
  const int kh = (lane >> 4) << 1;     // 0 or 2: K sub-offset per ISA A/B layout
  const int c0 = wave << 4;

  v8f acc0 = {}; v8f acc1 = {}; v8f acc2 = {}; v8f acc3 = {};
  for (int k0 = 0; k0 < KD; k0 += 4) {
    const int kb = k0 + kh;
    v2f b;
    b.x = W[(size_t)kb * HID + c0 + n];
    b.y = W[(size_t)(kb + 1) * HID + c0 + n];
    v2f a0, a1, a2, a3;
    a0.x = hs[n * AS + kb];        a0.y = hs[n * AS + kb + 1];
    a1.x = hs[(16 + n) * AS + kb]; a1.y = hs[(16 + n) * AS + kb + 1];
    a2.x = hs[(32 + n) * AS + kb]; a2.y = hs[(32 + n) * AS + kb + 1];
    a3.x = hs[(48 + n) * AS + kb]; a3.y = hs[(48 + n) * AS + kb + 1];
    acc0 = wmma_f32_k4(a0, b, acc0);
    acc1 = wmma_f32_k4(a1, b, acc1);
    acc2 = wmma_f32_k4(a2, b, acc2);
    acc3 = wmma_f32_k4(a3, b, acc3);
  }

  const float bv = bias ? bias[c0 + n] : 0.0f;
  const int lh = lane >> 4;
#pragma unroll
  for (int j = 0; j < 8; ++j) {
    int m = j + (lh << 3);             // C/D: VGPR j <-> row j + 8*(lane>>4)
    int r0 = rbase + m;
    if (r0 < nrows) out[(size_t)r0 * HID + c0 + n] = acc0[j] + bv;
    int r1 = rbase + 16 + m;
    if (r1 < nrows) out[(size_t)r1 * HID + c0 + n] = acc1[j] + bv;
    int r2 = rbase + 32 + m;
    if (r2 < nrows) out[(size_t)r2 * HID + c0 + n] = acc2[j] + bv;
    int r3 = rbase + 48 + m;
    if (r3 < nrows) out[(size_t)r3 * HID + c0 + n] = acc3[j] + bv;
  }
}

// ---------------- agg init: self-loop + bias ----------------

__global__ void init_agg_kernel(const float* __restrict__ m, const float* __restrict__ dis,
                                const float* __restrict__ bias, float* __restrict__ agg,
                                int total) {
  int idx = blockIdx.x * blockDim.x + threadIdx.x;
  if (idx >= total) return;
  int node = idx >> 7;
  int f = idx & 127;
  float di = dis[node];
  agg[idx] = m[idx] * di * di + bias[f];
}

// ---------------- edge scatter: one wave per edge ----------------

__global__ void scatter_edges_kernel(const long long* __restrict__ src,
                                     const long long* __restrict__ dst,
                                     const float* __restrict__ dis,
                                     const float* __restrict__ m,
                                     float* __restrict__ agg, int e) {
  int t = blockIdx.x * blockDim.x + threadIdx.x;
  int edge = t >> 5;
  int lane = t & 31;
  if (edge >= e) return;
  int s = (int)src[edge];
  int d = (int)dst[edge];
  float c = dis[s] * dis[d];
  float4 v = ((const float4*)(m + (size_t)s * HID))[lane];
  float* ap = agg + (size_t)d * HID + lane * 4;
  atomicAdd(ap + 0, v.x * c);
  atomicAdd(ap + 1, v.y * c);
  atomicAdd(ap + 2, v.z * c);
  atomicAdd(ap + 3, v.w * c);
}

// ---------------- LayerNorm + ReLU (+ residual): one wave per node ----------------

__global__ void ln_relu_kernel(const float* __restrict__ agg, const float* __restrict__ h_prev,
                               const float* __restrict__ gamma, const float* __restrict__ beta,
                               float* __restrict__ out, int n) {
  int t = blockIdx.x * blockDim.x + threadIdx.x;
  int node = t >> 5;
  int lane = t & 31;
  if (node >= n) return;
  float4 v = ((const float4*)(agg + (size_t)node * HID))[lane];
  float s  = v.x + v.y + v.z + v.w;
  float sq = v.x * v.x + v.y * v.y + v.z * v.z + v.w * v.w;
#pragma unroll
  for (int off = 16; off > 0; off >>= 1) {
    s  += __shfl_xor(s, off, 32);
    sq += __shfl_xor(sq, off, 32);
  }
  float mu  = s * (1.0f / HID);
  float var = sq * (1.0f / HID) - mu * mu;
  float rs  = rsqrtf(var + LN_EPS);
  float4 g = ((const float4*)gamma)[lane];
  float4 b = ((const float4*)beta)[lane];
  float4 r;
  r.x = fmaxf((v.x - mu) * rs * g.x + b.x, 0.0f);
  r.y = fmaxf((v.y - mu) * rs * g.y + b.y, 0.0f);
  r.z = fmaxf((v.z - mu) * rs * g.z + b.z, 0.0f);
  r.w = fmaxf((v.w - mu) * rs * g.w + b.w, 0.0f);
  if (h_prev) {
    float4 hp = ((const float4*)(h_prev + (size_t)node * HID))[lane];
    r.x += hp.x; r.y += hp.y; r.z += hp.z; r.w += hp.w;
  }
  ((float4*)(out + (size_t)node * HID))[lane] = r;
}

// ---------------- launch ----------------

extern "C" void kernel_launch(void* const* d_in, const int* in_sizes, int n_in,
                              void* d_out, int out_size, void* d_ws, size_t ws_size,
                              hipStream_t stream) {
  const float*     x      = (const float*)d_in[0];
  const long long* ei     = (const long long*)d_in[1];  // [2, E] int64
  const float*     W_in   = (const float*)d_in[2];
  const float*     b_in   = (const float*)d_in[3];
  const float*     W_conv = (const float*)d_in[4];      // [3,128,128]
  const float*     b_conv = (const float*)d_in[5];      // [3,128]
  const float*     ln_g   = (const float*)d_in[6];      // [3,128]
  const float*     ln_b   = (const float*)d_in[7];      // [3,128]
  float*           outp   = (float*)d_out;

  const int N = N_NODES, E = N_EDGES;
  const size_t HROW = (size_t)N * HID;

  char* ws = (char*)d_ws;
  float* dis = (float*)ws;
  size_t off = (((size_t)N * 4) + 255) & ~(size_t)255;
  float* hA = (float*)(ws + off);
  float* hB = (float*)(ws + off + HROW * 4);

  const long long* srcI = ei;
  const long long* dstI = ei + E;

  // 1) degrees -> dis = rsqrt(deg)
  deg_init_kernel<<<(N + 255) / 256, 256, 0, stream>>>(dis, N);
  deg_edges_kernel<<<(E + 255) / 256, 256, 0, stream>>>(dstI, dis, E);
  deg_to_dis_kernel<<<(N + 255) / 256, 256, 0, stream>>>(dis, N);

  // 2) input projection: hA = x @ W_in + b_in
  gemm_rowslab_kernel<IN_DIM><<<(N + 63) / 64, 256, 0, stream>>>(x, W_in, b_in, hA, N);

  // 3) three GCN layers, rotating {hA, hB, d_out}
  struct Bufs { const float* in; float* m; float* agg; const float* res; float* out; };
  Bufs L[3] = {
    { hA,   hB, outp, nullptr, outp },  // l=0: no residual
    { outp, hA, hB,   outp,    hA   },  // l=1
    { hA,   hB, outp, hA,      outp },  // l=2: final result -> d_out
  };

  const int totalF  = N * HID;
  const int edgeThr = E * 32;
  const int nodeThr = N * 32;
  for (int l = 0; l < 3; ++l) {
    gemm_rowslab_kernel<HID><<<(N + 63) / 64, 256, 0, stream>>>(
        L[l].in, W_conv + (size_t)l * HID * HID, nullptr, L[l].m, N);
    init_agg_kernel<<<(totalF + 255) / 256, 256, 0, stream>>>(
        L[l].m, dis, b_conv + l * HID, L[l].agg, totalF);
    scatter_edges_kernel<<<(edgeThr + 255) / 256, 256, 0, stream>>>(
        srcI, dstI, dis, L[l].m, L[l].agg, E);
    ln_relu_kernel<<<(nodeThr + 255) / 256, 256, 0, stream>>>(
        L[l].agg, L[l].res, ln_g + l * HID, ln_b + l * HID, L[l].out, N);
  }
}